// OctreeCrossEntropyLoss_55972013802040
// MI455X (gfx1250) — compile-verified
//
#include <hip/hip_runtime.h>
#include <math.h>

// OctreeCrossEntropyLoss for MI455X (gfx1250, wave32).
//
// Kernel 1: one workgroup (256 thr = 8 waves) per 16^3 block; streams gt_label
//           + dense_logits with vectorized b128 loads (each thread owns 4
//           consecutive voxels x 4 chunks; each wave covers a contiguous 512B
//           span -> perfectly coalesced global_load_b128). Per-voxel 2-class
//           NLL via softplus form (1 exp + 1 log1p). Each wave's 32 partials
//           are reduced deterministically with V_WMMA_F32_16X16X4_F32 (A =
//           lane partials, B = ones), then 8 wave sums combine in LDS. Also
//           emits per-block purity flags (bit0 = hasZero, bit1 = hasOne).
// Kernel 2: single workgroup; fixed-order tree reductions (deterministic) of
//           the 4096 block losses, hierarchical OR of flags for levels 1..4,
//           3-class CE terms scaled by b^3, writes scalar loss.

typedef __attribute__((ext_vector_type(2))) float v2f;
typedef __attribute__((ext_vector_type(8))) float v8f;

__global__ __launch_bounds__(256) void octree_level0_kernel(
    const int* __restrict__ gt,          // [256,256,256] +-1
    const float* __restrict__ logits,    // [4096, 2, 16,16,16]
    float* __restrict__ blockLoss,       // [4096]
    int* __restrict__ blockFlags)        // [4096] bit0=hasZero bit1=hasOne
{
  const int k   = blockIdx.x;            // block id = (bx*16 + by)*16 + bz
  const int tid = threadIdx.x;           // 0..255
  const int bx = k >> 8, by = (k >> 4) & 15, bz = k & 15;
  const float* lb = logits + (size_t)k * 8192;

  float acc = 0.0f;
  int hasZ = 0, hasO = 0;
#pragma unroll
  for (int q = 0; q < 4; ++q) {
    // 4 consecutive inner indices j..j+3 (16B aligned); wave-contiguous.
    const int j  = q * 1024 + tid * 4;
    const int ix = j >> 8;               // inner x
    const int iy = (j >> 4) & 15;        // inner y
    const int iz = j & 15;               // inner z, multiple of 4
    const int x = (bx << 4) + ix;
    const int y = (by << 4) + iy;
    const int z = (bz << 4) + iz;

    const int4   g4 = *(const int4*)  (gt + (((x << 8) + y) << 8) + z);
    const float4 a0 = *(const float4*)(lb + j);
    const float4 a1 = *(const float4*)(lb + 4096 + j);

    const int   gs[4]  = {g4.x, g4.y, g4.z, g4.w};
    const float l0s[4] = {a0.x, a0.y, a0.z, a0.w};
    const float l1s[4] = {a1.x, a1.y, a1.z, a1.w};
#pragma unroll
    for (int u = 0; u < 4; ++u) {
      const float l0 = l0s[u];
      const float l1 = l1s[u];
      const int t = (gs[u] > 0) ? 1 : 0;       // {-1,+1} -> {0,1}
      hasO |= t;
      hasZ |= (t ^ 1);
      // nll = logsumexp(l0,l1) - l_t = (max - l_t) + log1p(exp(-|l0-l1|))
      const float m  = fmaxf(l0, l1);
      const float lt = t ? l1 : l0;
      acc += (m - lt) + log1pf(expf(-fabsf(l0 - l1)));
    }
  }

  // ---- deterministic intra-wave sum of 32 partials via WMMA f32 16x16x4 ----
  // A layout (16x4 f32): lanes 0-15 hold A[L][0],A[L][1]; lanes 16-31 hold
  // A[L-16][2],A[L-16][3]. With a.y=0 and B = all-ones:
  //   D[m][n] = p_m + p_{m+16}  (independent of n)
  // Per-lane sum of D's 8 VGPRs gives sum(rows 0..7) on lanes <16 and
  // sum(rows 8..15) on lanes >=16; lane0 + lane16 = full wave sum.
  v2f a; a.x = acc;  a.y = 0.0f;
  v2f b; b.x = 1.0f; b.y = 1.0f;
  v8f c = {};
  v8f d = __builtin_amdgcn_wmma_f32_16x16x4_f32(
      /*neg_a=*/false, a, /*neg_b=*/false, b,
      /*c_mod=*/(short)0, c, /*reuse_a=*/false, /*reuse_b=*/false);
  const float s = d[0] + d[1] + d[2] + d[3] + d[4] + d[5] + d[6] + d[7];
  const float waveTot = __shfl(s, 0, 32) + __shfl(s, 16, 32);

  const int wz = __any(hasZ);
  const int wo = __any(hasO);

  __shared__ float wsum[8];
  __shared__ int   wflag[8];
  const int wave = tid >> 5;
  if ((tid & 31) == 0) {
    wsum[wave]  = waveTot;
    wflag[wave] = (wz ? 1 : 0) | (wo ? 2 : 0);
  }
  __syncthreads();
  if (tid == 0) {
    float tot = 0.0f;
    int f = 0;
    for (int w = 0; w < 8; ++w) { tot += wsum[w]; f |= wflag[w]; }
    blockLoss[k]  = tot * (1.0f / 4096.0f);   // mean over voxels
    blockFlags[k] = f;
  }
}

// Fixed-order tree reduction over 256 lanes; bitwise deterministic.
__device__ __forceinline__ float tree_reduce256(float* red, float v, int tid) {
  red[tid] = v;
  __syncthreads();
  for (int s = 128; s > 0; s >>= 1) {
    if (tid < s) red[tid] += red[tid + s];
    __syncthreads();
  }
  const float r = red[0];
  __syncthreads();
  return r;
}

__device__ __forceinline__ float nll3(const float* __restrict__ lg, int cls) {
  const float a = lg[0], b = lg[1], c = lg[2];
  const float m = fmaxf(a, fmaxf(b, c));
  const float lse = m + logf(expf(a - m) + expf(b - m) + expf(c - m));
  return lse - lg[cls];
}

__device__ __forceinline__ int cls_from_flags(int f) {
  // f in {1,2,3}: 1 -> all-zero (cls 0), 2 -> all-one (cls 1), 3 -> mixed (2)
  return (f == 3) ? 2 : (f >> 1);
}

__global__ __launch_bounds__(256) void octree_finalize_kernel(
    const float* __restrict__ blockLoss,
    const int* __restrict__ blockFlags,
    const float* __restrict__ l1,   // [512,3]
    const float* __restrict__ l2,   // [64,3]
    const float* __restrict__ l3,   // [8,3]
    const float* __restrict__ l4,   // [1,3]
    float* __restrict__ out)
{
  __shared__ float red[256];
  __shared__ int f1s[512];
  __shared__ int f2s[64];
  __shared__ int f3s[8];
  const int tid = threadIdx.x;

  // ---- level 0: sum 4096 block losses, fixed order ----
  float p = 0.0f;
  for (int i = 0; i < 16; ++i) p += blockLoss[tid * 16 + i];
  float total = tree_reduce256(red, p, tid);

  // ---- level 1: 8^3 = 512 coarse blocks of 32^3 ----
  float p1 = 0.0f;
  for (int K = tid; K < 512; K += 256) {
    const int cx = K >> 6, cy = (K >> 3) & 7, cz = K & 7;
    int f = 0;
    for (int dd = 0; dd < 8; ++dd) {
      const int fx = 2 * cx + (dd >> 2);
      const int fy = 2 * cy + ((dd >> 1) & 1);
      const int fz = 2 * cz + (dd & 1);
      f |= blockFlags[(fx << 8) + (fy << 4) + fz];
    }
    f1s[K] = f;
    p1 += 32768.0f * nll3(l1 + K * 3, cls_from_flags(f));   // 32^3
  }
  total += tree_reduce256(red, p1, tid);

  // ---- level 2: 4^3 = 64 blocks of 64^3 ----
  float p2 = 0.0f;
  if (tid < 64) {
    const int K = tid;
    const int cx = K >> 4, cy = (K >> 2) & 3, cz = K & 3;
    int f = 0;
    for (int dd = 0; dd < 8; ++dd) {
      const int fx = 2 * cx + (dd >> 2);
      const int fy = 2 * cy + ((dd >> 1) & 1);
      const int fz = 2 * cz + (dd & 1);
      f |= f1s[(fx << 6) + (fy << 3) + fz];
    }
    f2s[K] = f;
    p2 = 262144.0f * nll3(l2 + K * 3, cls_from_flags(f));   // 64^3
  }
  total += tree_reduce256(red, p2, tid);

  // ---- level 3: 2^3 = 8 blocks of 128^3 ----
  float p3 = 0.0f;
  if (tid < 8) {
    const int K = tid;
    const int cx = K >> 2, cy = (K >> 1) & 1, cz = K & 1;
    int f = 0;
    for (int dd = 0; dd < 8; ++dd) {
      const int fx = 2 * cx + (dd >> 2);
      const int fy = 2 * cy + ((dd >> 1) & 1);
      const int fz = 2 * cz + (dd & 1);
      f |= f2s[(fx << 4) + (fy << 2) + fz];
    }
    f3s[K] = f;
    p3 = 2097152.0f * nll3(l3 + K * 3, cls_from_flags(f));  // 128^3
  }
  total += tree_reduce256(red, p3, tid);                    // also a barrier

  // ---- level 4: single 256^3 block ----
  if (tid == 0) {
    int f = 0;
    for (int dd = 0; dd < 8; ++dd) f |= f3s[dd];
    total += 16777216.0f * nll3(l4, cls_from_flags(f));     // 256^3
    out[0] = total;
  }
}

extern "C" void kernel_launch(void* const* d_in, const int* in_sizes, int n_in,
                              void* d_out, int out_size, void* d_ws, size_t ws_size,
                              hipStream_t stream) {
  (void)in_sizes; (void)n_in; (void)out_size; (void)ws_size;
  const int*   gt    = (const int*)d_in[0];     // gt_label [1,256,256,256]
  const float* dense = (const float*)d_in[1];   // dense_logits [4096,2,16,16,16]
  const float* l1    = (const float*)d_in[2];   // [512,3]
  const float* l2    = (const float*)d_in[3];   // [64,3]
  const float* l3    = (const float*)d_in[4];   // [8,3]
  const float* l4    = (const float*)d_in[5];   // [1,3]

  float* blockLoss  = (float*)d_ws;                                  // 4096 f32
  int*   blockFlags = (int*)((char*)d_ws + 4096 * sizeof(float));    // 4096 i32

  octree_level0_kernel<<<4096, 256, 0, stream>>>(gt, dense, blockLoss, blockFlags);
  octree_finalize_kernel<<<1, 256, 0, stream>>>(blockLoss, blockFlags,
                                                l1, l2, l3, l4, (float*)d_out);
}